// ABCNN3_73254962200864
// MI455X (gfx1250) — compile-verified
//
#include <hip/hip_runtime.h>
#include <hip/hip_bf16.h>
#include <math.h>

// ABCNN3 pipeline for MI455X (gfx1250, wave32).
// Conv1d(30->30,K=4,pad=3) expressed as K=120 GEMM contraction executed as
// 30 chained V_WMMA_F32_16X16X4_F32 per 16x16 output tile. Entire sentence
// (plus conv intermediates + both weight sets) kept resident in WGP LDS.
// A-operands (weights) hoisted to registers; two independent accumulator
// chains per wave to hide WMMA latency; weights zero-padded to 32 rows so
// the inner loop is branch/exec-mask free.

typedef __attribute__((ext_vector_type(2))) float v2f;
typedef __attribute__((ext_vector_type(8))) float v8f;

#define EMB   30
#define KW    4
#define CKD   (EMB*KW)            // 120 contraction length
#define NSENT 1024
#define SLEN  512
#define QLEN  128
#define XS    544                 // LDS row stride (floats), covers 33 tiles + taps
#define WSZ   (EMB*CKD)           // 3600 real weights per conv layer
#define WPAD  (32*CKD)            // 3840 padded (rows 30,31 = zero)
#define LDS_FLOATS (2*EMB*XS + 2*WPAD + 64 + 32)

__device__ __forceinline__ void block_zero(float* p, int n) {
  for (int i = threadIdx.x; i < n; i += blockDim.x) p[i] = 0.f;
}

// sums buf[c*stride + off + t] over t in [0,len), writes mean to outp[c], c<30
__device__ void reduce_mean(const float* buf, int stride, int off, int len,
                            float* sums, float* outp) {
  const int tid = threadIdx.x;
  if (tid < 32) sums[tid] = 0.f;
  __syncthreads();
  if (tid < 240) {                 // 8 partial threads per channel
    const int c = tid % EMB, sub = tid / EMB;
    float partial = 0.f;
    for (int t = sub; t < len; t += 8) partial += buf[c * stride + off + t];
    atomicAdd(&sums[c], partial);  // ds_add_f32
  }
  __syncthreads();
  if (tid < EMB) outp[tid] = sums[tid] / (float)len;
  __syncthreads();
}

__device__ __forceinline__ void store_tile(float* __restrict__ ybuf,
                                           const float* __restrict__ bl,
                                           v8f acc, int ct, int pt,
                                           int col, int hi, int Lout) {
  const int t = pt * 16 + col;                  // D col -> output position
  if (t < Lout) {
#pragma unroll
    for (int r = 0; r < 8; ++r) {               // D vgpr r -> row r + 8*hi
      const int cc = ct * 16 + r + 8 * hi;
      if (cc < EMB) ybuf[cc * XS + t] = acc[r] + bl[cc];
    }
  }
}

// y[o,t] = bias[o] + sum_{i,k} w[o,i,k] * xpad[i, t+k]   (xpad = x with 3 left zeros)
// wl32: weights padded to 32 output rows (rows 30,31 zero) -> no guards on A loads.
__device__ void conv_wmma(const float* __restrict__ wl32, const float* __restrict__ bl,
                          const float* __restrict__ xbuf, float* __restrict__ ybuf,
                          int Lout) {
  const int lane = threadIdx.x & 31;
  const int wave = threadIdx.x >> 5;
  const int nwav = blockDim.x >> 5;
  const int col  = lane & 15;      // A: row M / B: col N / D: col N
  const int hi   = lane >> 4;      // lane half selects K pair (A/B), +8 rows (D)
  const int krow = hi << 1;        // K base for this lane half
  const int ntile = (Lout + 15) >> 4;

  for (int ct = 0; ct < 2; ++ct) {
    // ---- hoist A: all 30 contraction-step weight pairs for this channel tile
    const int ch = ct * 16 + col;                // 0..31, padded rows are zero
    v2f a[EMB];
#pragma unroll
    for (int j = 0; j < EMB; ++j) {
      a[j][0] = wl32[ch * CKD + j * KW + krow];
      a[j][1] = wl32[ch * CKD + j * KW + krow + 1];
    }

    int pt = wave;
    // ---- dual-tile main loop: two independent WMMA accumulator chains
    for (; pt + nwav < ntile; pt += 2 * nwav) {
      const int x0 = pt * 16 + col + krow;
      const int x1 = (pt + nwav) * 16 + col + krow;
      v8f acc0 = {0.f, 0.f, 0.f, 0.f, 0.f, 0.f, 0.f, 0.f};
      v8f acc1 = {0.f, 0.f, 0.f, 0.f, 0.f, 0.f, 0.f, 0.f};
#pragma unroll
      for (int j = 0; j < EMB; ++j) {
        v2f b0, b1;
        b0[0] = xbuf[j * XS + x0]; b0[1] = xbuf[j * XS + x0 + 1];
        b1[0] = xbuf[j * XS + x1]; b1[1] = xbuf[j * XS + x1 + 1];
        acc0 = __builtin_amdgcn_wmma_f32_16x16x4_f32(false, a[j], false, b0,
                                                     (short)0, acc0, false, false);
        acc1 = __builtin_amdgcn_wmma_f32_16x16x4_f32(false, a[j], false, b1,
                                                     (short)0, acc1, false, false);
      }
      store_tile(ybuf, bl, acc0, ct, pt, col, hi, Lout);
      store_tile(ybuf, bl, acc1, ct, pt + nwav, col, hi, Lout);
    }
    // ---- remainder tile
    for (; pt < ntile; pt += nwav) {
      const int x0 = pt * 16 + col + krow;
      v8f acc0 = {0.f, 0.f, 0.f, 0.f, 0.f, 0.f, 0.f, 0.f};
#pragma unroll
      for (int j = 0; j < EMB; ++j) {
        v2f b0;
        b0[0] = xbuf[j * XS + x0]; b0[1] = xbuf[j * XS + x0 + 1];
        acc0 = __builtin_amdgcn_wmma_f32_16x16x4_f32(false, a[j], false, b0,
                                                     (short)0, acc0, false, false);
      }
      store_tile(ybuf, bl, acc0, ct, pt, col, hi, Lout);
    }
  }
}

__global__ __launch_bounds__(256) void abcnn_pipeline(
    const float* __restrict__ q, const float* __restrict__ s,
    const float* __restrict__ w1, const float* __restrict__ b1,
    const float* __restrict__ w2, const float* __restrict__ b2,
    float* __restrict__ pools) {
  extern __shared__ float lds[];
  float* xbuf = lds;                       // 30*544  zero-padded input / window pool
  float* ybuf = lds + EMB * XS;            // 30*544  conv output
  float* wlds = lds + 2 * EMB * XS;        // 2*3840  padded conv1 + conv2 weights
  float* blds = wlds + 2 * WPAD;           // 64      biases
  float* sums = blds + 64;                 // 32      reduction scratch

  const int  bid  = blockIdx.x;
  const bool isQ  = (bid == NSENT);
  const int  L    = isQ ? QLEN : SLEN;
  const int  Lout = L + 3;                 // pad=3, kernel=4
  const float* x  = isQ ? q : (s + (size_t)bid * SLEN * EMB);
  float* pw = pools + (size_t)bid * 96;    // [raw|g1|g2] x 32

  block_zero(xbuf, 2 * EMB * XS + 2 * WPAD);   // also zeroes padded weight rows
  __syncthreads();
  for (int i = threadIdx.x; i < WSZ; i += blockDim.x) {
    wlds[i] = w1[i];                       // rows 0..29 (padded layout is identity
    wlds[WPAD + i] = w2[i];                //  for the first 3600 entries)
  }
  if (threadIdx.x < EMB) {
    blds[threadIdx.x] = b1[threadIdx.x];
    blds[32 + threadIdx.x] = b2[threadIdx.x];
  }
  // coalesced global read [t][c] -> LDS [c][3+t]
  for (int i = threadIdx.x; i < L * EMB; i += blockDim.x) {
    const int t = i / EMB, c = i % EMB;
    xbuf[c * XS + 3 + t] = x[i];
  }
  __syncthreads();

  reduce_mean(xbuf, XS, 3, L, sums, pw);                 // raw global pool
  conv_wmma(wlds, blds, xbuf, ybuf, Lout);               // conv1 (WMMA)
  __syncthreads();
  reduce_mean(ybuf, XS, 0, Lout, sums, pw + 32);         // level-1 pool

  // window pool (avg4, stride1): y[t..t+3] -> xbuf (pads stay zero)
  for (int i = threadIdx.x; i < EMB * L; i += blockDim.x) {
    const int c = i / L, t = i % L;
    const float* yy = ybuf + c * XS + t;
    xbuf[c * XS + 3 + t] = 0.25f * (yy[0] + yy[1] + yy[2] + yy[3]);
  }
  __syncthreads();

  conv_wmma(wlds + WPAD, blds + 32, xbuf, ybuf, Lout);   // conv2 (WMMA)
  __syncthreads();
  reduce_mean(ybuf, XS, 0, Lout, sums, pw + 64);         // level-2 pool
}

__global__ __launch_bounds__(256) void abcnn_head(
    const float* __restrict__ pools, const float* __restrict__ gaf,
    const int* __restrict__ labels, const float* __restrict__ lw,
    const float* __restrict__ lb, float* __restrict__ out) {
  __shared__ float qv[96];
  __shared__ float qn[3];
  __shared__ float costsum;
  const int tid = threadIdx.x;
  if (tid < 96) qv[tid] = pools[(size_t)NSENT * 96 + tid];
  if (tid == 0) costsum = 0.f;
  __syncthreads();
  if (tid < 3) {
    float ss = 0.f;
    for (int c = 0; c < EMB; ++c) { const float v = qv[tid * 32 + c]; ss += v * v; }
    qn[tid] = sqrtf(ss);
  }
  __syncthreads();

  float lpart = 0.f;
  for (int n = tid; n < NSENT; n += blockDim.x) {
    const float* sp = pools + (size_t)n * 96;
    const float* qsel = (n == 0) ? qv : (qv + 64);   // source-loop aliasing quirk
    float d1 = 0.f, d2 = 0.f, d3 = 0.f, nr = 0.f, n1 = 0.f, n2 = 0.f;
    for (int c = 0; c < EMB; ++c) {
      const float sr = sp[c], s1 = sp[32 + c], s2 = sp[64 + c];
      d1 += sr * qsel[c];  d2 += s1 * qv[32 + c];  d3 += s2 * qv[64 + c];
      nr += sr * sr;       n1 += s1 * s1;          n2 += s2 * s2;
    }
    const float qns = (n == 0) ? qn[0] : qn[2];
    const float sim1 = d1 / (sqrtf(nr) * qns);
    const float sim2 = d2 / (sqrtf(n1) * qn[1]);
    const float sim3 = d3 / (sqrtf(n2) * qn[2]);
    float z0 = lb[0] + lw[0] * sim1 + lw[1] * sim2 + lw[2] * sim3;
    float z1 = lb[1] + lw[12] * sim1 + lw[13] * sim2 + lw[14] * sim3;
    for (int k = 0; k < 9; ++k) {
      const float g = gaf[n * 9 + k];
      z0 += lw[3 + k] * g;
      z1 += lw[15 + k] * g;
    }
    const float m = fmaxf(z0, z1);
    const float lse = m + logf(expf(z0 - m) + expf(z1 - m));
    const float lp0 = z0 - lse, lp1 = z1 - lse;
    const float e0 = expf(lp0), e1 = expf(lp1);     // softmax(log_softmax), as source
    out[1 + n] = e1 / (e0 + e1);
    lpart += -((labels[n] != 0) ? lp1 : lp0);
  }
  atomicAdd(&costsum, lpart);
  __syncthreads();
  if (tid == 0) out[0] = costsum / (float)NSENT;
}

extern "C" void kernel_launch(void* const* d_in, const int* in_sizes, int n_in,
                              void* d_out, int out_size, void* d_ws, size_t ws_size,
                              hipStream_t stream) {
  (void)in_sizes; (void)n_in; (void)out_size; (void)ws_size;
  const float* q   = (const float*)d_in[0];
  const float* s   = (const float*)d_in[1];
  const float* gaf = (const float*)d_in[2];
  const int*   lab = (const int*)d_in[3];
  const float* w1  = (const float*)d_in[4];
  const float* b1  = (const float*)d_in[5];
  const float* w2  = (const float*)d_in[6];
  const float* b2  = (const float*)d_in[7];
  const float* lw  = (const float*)d_in[8];
  const float* lb  = (const float*)d_in[9];
  float* out   = (float*)d_out;
  float* pools = (float*)d_ws;   // (1024+1) * 96 floats

  const size_t ldsB = (size_t)LDS_FLOATS * sizeof(float);  // ~158 KB (<320 KB WGP)
  hipFuncSetAttribute(reinterpret_cast<const void*>(abcnn_pipeline),
                      hipFuncAttributeMaxDynamicSharedMemorySize, (int)ldsB);
  abcnn_pipeline<<<NSENT + 1, 256, ldsB, stream>>>(q, s, w1, b1, w2, b2, pools);
  abcnn_head<<<1, 256, 0, stream>>>(pools, gaf, lab, lw, lb, out);
}